// GAT_29437705846970
// MI455X (gfx1250) — compile-verified
//
#include <hip/hip_runtime.h>

typedef __attribute__((ext_vector_type(2))) float v2f;
typedef __attribute__((ext_vector_type(8))) float v8f;

#define NEG_SLOPE 0.2f

// ---- ordered-uint encoding for float atomic max (handles negatives) ----
__device__ __forceinline__ unsigned fenc(float x) {
  unsigned u = __float_as_uint(x);
  return (u & 0x80000000u) ? ~u : (u | 0x80000000u);
}
__device__ __forceinline__ float fdec(unsigned u) {
  unsigned v = (u & 0x80000000u) ? (u & 0x7FFFFFFFu) : ~u;
  return __uint_as_float(v);
}

// ============================================================
// Dual GEMM: H = X@W, R = X@Rw using V_WMMA_F32_16X16X4_F32.
// One wave -> one 16x16 tile of both outputs; A tile feeds two WMMAs.
// DI/DN compile-time so all load offsets become immediates.
// ============================================================
template <int DI, int DN>
__global__ __launch_bounds__(256) void gat_gemm_dual_t(
    const float* __restrict__ X, const float* __restrict__ W,
    const float* __restrict__ Rw, float* __restrict__ H,
    float* __restrict__ Rout, int tiles) {
  int wv   = threadIdx.x >> 5;
  int lane = threadIdx.x & 31;
  int tile = blockIdx.x * 8 + wv;
  if (tile >= tiles) return;                 // wave-uniform: EXEC stays all-1s
  constexpr int colTiles = DN >> 4;
  int tm = tile / colTiles;
  int tn = tile - tm * colTiles;
  int half = lane >> 4;                      // K-half select (ISA A/B layout)
  int l15  = lane & 15;
  int col  = tn * 16 + l15;
  const float* xb = X  + (size_t)(tm * 16 + l15) * DI + (half << 1);
  const float* wb = W  + (size_t)(half << 1) * DN + col;
  const float* rb = Rw + (size_t)(half << 1) * DN + col;
  v8f acch = {};
  v8f accr = {};
#pragma unroll 8
  for (int k = 0; k < DI; k += 4) {
    v2f a = *(const v2f*)(xb + k);           // K = base, base+1 of this lane's row
    v2f bw, br;
    bw.x = wb[k * DN]; bw.y = wb[k * DN + DN];
    br.x = rb[k * DN]; br.y = rb[k * DN + DN];
    acch = __builtin_amdgcn_wmma_f32_16x16x4_f32(false, a, false, bw, (short)0, acch, false, false);
    accr = __builtin_amdgcn_wmma_f32_16x16x4_f32(false, a, false, br, (short)0, accr, false, false);
  }
  float* hp = H    + (size_t)(tm * 16 + (half << 3)) * DN + col;
  float* rp = Rout + (size_t)(tm * 16 + (half << 3)) * DN + col;
#pragma unroll
  for (int v = 0; v < 8; ++v) {              // C/D layout: VGPR v -> row v + 8*half
    hp[v * DN] = acch[v];
    rp[v * DN] = accr[v];
  }
}

// ============================================================
// Per-node attention logits: as[n]=h[n]·a_s, ad[n]=h[n]·a_d
// One wave32 per node, butterfly reduction.
// ============================================================
template <int DN>
__global__ __launch_bounds__(256) void gat_alpha(
    const float* __restrict__ H, const float* __restrict__ a_s,
    const float* __restrict__ a_d, float* __restrict__ AS,
    float* __restrict__ AD, int Nn) {
  int gw   = (int)((blockIdx.x * blockDim.x + threadIdx.x) >> 5);
  int lane = threadIdx.x & 31;
  if (gw >= Nn) return;
  const float* hr = H + (size_t)gw * DN;
  float ps = 0.f, pd = 0.f;
#pragma unroll
  for (int f0 = 0; f0 < DN; f0 += 32) {
    int f = f0 + lane;
    float hv = hr[f];
    ps += hv * a_s[f];
    pd += hv * a_d[f];
  }
#pragma unroll
  for (int off = 16; off > 0; off >>= 1) {
    ps += __shfl_xor(ps, off, 32);
    pd += __shfl_xor(pd, off, 32);
  }
  if (lane == 0) { AS[gw] = ps; AD[gw] = pd; }
}

// ============================================================
// Init accumulators: agg=0, menc=0 (== below any float in ordered-uint), ssum=0
// ============================================================
__global__ __launch_bounds__(256) void gat_init(
    unsigned* __restrict__ menc, float* __restrict__ ss,
    float* __restrict__ ag, int Nn, int total) {
  int i = blockIdx.x * blockDim.x + threadIdx.x;
  if (i < total) ag[i] = 0.f;
  if (i < Nn) { menc[i] = 0u; ss[i] = 0.f; }
}

// ============================================================
// Edge pass 1: e = leaky_relu(as[src]+ad[dst]); segment-max via atomicMax
// ============================================================
__global__ __launch_bounds__(256) void gat_edge_max(
    const int* __restrict__ ei, const float* __restrict__ AS,
    const float* __restrict__ AD, float* __restrict__ elog,
    unsigned* __restrict__ menc, int E_, int Nn) {
  int e = blockIdx.x * blockDim.x + threadIdx.x;
  int EE = E_ + Nn;
  if (e >= EE) return;
  int s, d;
  if (e < E_) { s = ei[e]; d = ei[E_ + e]; } else { s = d = e - E_; }
  float v = AS[s] + AD[d];
  v = (v > 0.f) ? v : NEG_SLOPE * v;
  elog[e] = v;
  atomicMax(menc + d, fenc(v));
}

// ============================================================
// Edge pass 2: w = exp(e - m[dst]); segment-sum via atomicAdd
// ============================================================
__global__ __launch_bounds__(256) void gat_edge_sum(
    const int* __restrict__ ei, const float* __restrict__ elog,
    const unsigned* __restrict__ menc, float* __restrict__ wbuf,
    float* __restrict__ ss, int E_, int Nn) {
  int e = blockIdx.x * blockDim.x + threadIdx.x;
  int EE = E_ + Nn;
  if (e >= EE) return;
  int d = (e < E_) ? ei[E_ + e] : (e - E_);
  float w = __expf(elog[e] - fdec(menc[d]));
  wbuf[e] = w;
  atomicAdd(ss + d, w);
}

// ============================================================
// Edge pass 3: agg[dst] += h[src] * (w / s[dst]); one wave32 per edge
// ============================================================
template <int DN>
__global__ __launch_bounds__(256) void gat_edge_agg(
    const int* __restrict__ ei, const float* __restrict__ H,
    const float* __restrict__ wbuf, const float* __restrict__ ss,
    float* __restrict__ ag, int E_, int Nn) {
  int gw   = (int)((blockIdx.x * blockDim.x + threadIdx.x) >> 5);
  int lane = threadIdx.x & 31;
  int EE = E_ + Nn;
  if (gw >= EE) return;
  int s, d;
  if (gw < E_) { s = ei[gw]; d = ei[E_ + gw]; } else { s = d = gw - E_; }
  float alpha = wbuf[gw] / ss[d];
  const float* hs = H + (size_t)s * DN + lane;
  float* agp = ag + (size_t)d * DN + lane;
#pragma unroll
  for (int f = 0; f < DN; f += 32)
    atomicAdd(agp + f, hs[f] * alpha);
}

// ============================================================
// Finalize: x_next = relu(agg) + x@R
// ============================================================
__global__ __launch_bounds__(256) void gat_finalize(
    const float* __restrict__ ag, const float* __restrict__ R,
    float* __restrict__ xout, int total) {
  int i = blockIdx.x * blockDim.x + threadIdx.x;
  if (i >= total) return;
  float v = ag[i];
  v = (v > 0.f) ? v : 0.f;
  xout[i] = v + R[i];
}

extern "C" void kernel_launch(void* const* d_in, const int* in_sizes, int n_in,
                              void* d_out, int out_size, void* d_ws, size_t ws_size,
                              hipStream_t stream) {
  (void)n_in; (void)out_size; (void)ws_size;
  const float* x0 = (const float*)d_in[0];
  const int*   ei = (const int*)d_in[1];
  const int N  = in_sizes[0] / 256;
  const int E  = in_sizes[1] / 2;
  const int EE = E + N;

  static const int DOUT[4] = {96, 96, 96, 48};

  // workspace layout (floats)
  float*    f    = (float*)d_ws;
  size_t    N96  = (size_t)N * 96;
  float*    H    = f;                 // N*96
  float*    R    = H + N96;           // N*96
  float*    AG   = R + N96;           // N*96
  float*    XB   = AG + N96;          // N*96  (layer-output ping buffer)
  float*    AS   = XB + N96;          // N
  float*    AD   = AS + N;            // N
  float*    SS   = AD + N;            // N
  unsigned* MENC = (unsigned*)(SS + N);        // N
  float*    ELOG = (float*)(MENC + N);         // EE
  float*    WBUF = ELOG + EE;                  // EE

  const float* xcur = x0;
  for (int L = 0; L < 4; ++L) {
    const int dd = DOUT[L];
    const float* W  = (const float*)d_in[2 + 4 * L + 0];
    const float* As = (const float*)d_in[2 + 4 * L + 1];
    const float* Ad = (const float*)d_in[2 + 4 * L + 2];
    const float* Rw = (const float*)d_in[2 + 4 * L + 3];

    const int tiles = (N / 16) * (dd / 16);
    const int gb = (tiles + 7) / 8;
    switch (L) {
      case 0: gat_gemm_dual_t<256, 96><<<gb, 256, 0, stream>>>(xcur, W, Rw, H, R, tiles); break;
      case 1:
      case 2: gat_gemm_dual_t< 96, 96><<<gb, 256, 0, stream>>>(xcur, W, Rw, H, R, tiles); break;
      default: gat_gemm_dual_t< 96, 48><<<gb, 256, 0, stream>>>(xcur, W, Rw, H, R, tiles); break;
    }

    if (dd == 96) gat_alpha<96><<<(N + 7) / 8, 256, 0, stream>>>(H, As, Ad, AS, AD, N);
    else          gat_alpha<48><<<(N + 7) / 8, 256, 0, stream>>>(H, As, Ad, AS, AD, N);

    const int tot = N * dd;
    gat_init<<<(tot + 255) / 256, 256, 0, stream>>>(MENC, SS, AG, N, tot);

    gat_edge_max<<<(EE + 255) / 256, 256, 0, stream>>>(ei, AS, AD, ELOG, MENC, E, N);
    gat_edge_sum<<<(EE + 255) / 256, 256, 0, stream>>>(ei, ELOG, MENC, WBUF, SS, E, N);
    if (dd == 96) gat_edge_agg<96><<<(EE + 7) / 8, 256, 0, stream>>>(ei, H, WBUF, SS, AG, E, N);
    else          gat_edge_agg<48><<<(EE + 7) / 8, 256, 0, stream>>>(ei, H, WBUF, SS, AG, E, N);

    float* xnext = (L == 3) ? (float*)d_out : XB;
    gat_finalize<<<(tot + 255) / 256, 256, 0, stream>>>(AG, R, xnext, tot);
    xcur = xnext;
  }
}